// ContextComputer_26783416058386
// MI455X (gfx1250) — compile-verified
//
#include <hip/hip_runtime.h>

typedef float v2f __attribute__((ext_vector_type(2)));
typedef float v8f __attribute__((ext_vector_type(8)));

#define DDIM 512
#define NROW 6
#define GB   8                 // batches per workgroup
#define ROWS (GB * NROW)       // 48 rows of the flattened (B*N, D) activation
#define MT   (ROWS / 16)       // 3 WMMA M-tiles
#define SM   516               // padded LDS row stride (f32): 2064B, 16B-aligned, conflict-free
#define NTHREADS 512           // 16 waves (wave32)

__global__ __launch_bounds__(NTHREADS, 1)
void context_fused_kernel(const float* __restrict__ mem_g,
                          const float* __restrict__ mask_g,
                          const float* __restrict__ W_g,
                          const float* __restrict__ b_g,
                          float* __restrict__ out_g)
{
    __shared__ float s_mem[ROWS * SM];   // memory tile (48 x 512)
    __shared__ float s_a[ROWS * SM];     // a = mem @ W1
    __shared__ float s_c[ROWS * SM];     // c = (mem*mask) @ W2
    __shared__ float s_bias[DDIM];
    __shared__ float s_mask[ROWS];

    const int tid   = threadIdx.x;
    const int lane  = tid & 31;
    const int wave  = tid >> 5;          // 0..15
    const int laneM = lane & 15;         // M (A-frag) / N (B,D-frag) index
    const int laneH = lane >> 4;         // K-half selector

    const long rowBase = (long)blockIdx.x * ROWS;   // flattened (b*N + i) row base

    // ------- stage memory tile -> LDS via CDNA5 async copy (ASYNCcnt path) -------
    {
        // low 32 bits of a generic LDS pointer == workgroup-relative LDS byte offset
        const unsigned lds_mem_base = (unsigned)(unsigned long long)(uintptr_t)(void*)s_mem;
        const float* gbase = mem_g + rowBase * DDIM;
        #pragma unroll
        for (int it = 0; it < (ROWS * DDIM / 4) / NTHREADS; ++it) {   // 12 iters
            int idx = tid + it * NTHREADS;
            int r   = idx >> 7;          // / (512/4)
            int c4  = idx & 127;
            const float* gp = gbase + r * DDIM + c4 * 4;
            unsigned lds_addr = lds_mem_base + (unsigned)(r * SM + c4 * 4) * 4u;
            asm volatile("global_load_async_to_lds_b128 %0, %1, off"
                         :: "v"(lds_addr), "v"(gp) : "memory");
        }
        if (tid < ROWS) s_mask[tid] = mask_g[rowBase + tid];
        if (tid < DDIM) s_bias[tid] = b_g[tid];
        asm volatile("s_wait_asynccnt 0x0" ::: "memory");
    }
    __syncthreads();

    // ---------------- phase 1: dual GEMM via V_WMMA_F32_16X16X4_F32 --------
    const int colBase  = wave * 32;           // this wave owns 32 output columns
    const float* W1 = W_g;
    const float* W2 = W_g + DDIM * DDIM;

    float mv[MT];
    #pragma unroll
    for (int m = 0; m < MT; ++m) mv[m] = s_mask[m * 16 + laneM];

    v8f accA[MT][2], accC[MT][2];
    #pragma unroll
    for (int m = 0; m < MT; ++m)
        #pragma unroll
        for (int nt = 0; nt < 2; ++nt)
            #pragma unroll
            for (int q = 0; q < 8; ++q) { accA[m][nt][q] = 0.f; accC[m][nt][q] = 0.f; }

    for (int k = 0; k < DDIM; k += 4) {
        const int k0 = k + 2 * laneH;         // this lane's K pair: k0, k0+1

        // B-fragments for W1/W2 (K x 16 cols), read from L2-resident W
        v2f bf1[2], bf2[2];
        #pragma unroll
        for (int nt = 0; nt < 2; ++nt) {
            const int col = colBase + nt * 16 + laneM;
            bf1[nt].x = W1[(k0    ) * DDIM + col];
            bf1[nt].y = W1[(k0 + 1) * DDIM + col];
            bf2[nt].x = W2[(k0    ) * DDIM + col];
            bf2[nt].y = W2[(k0 + 1) * DDIM + col];
        }

        #pragma unroll
        for (int m = 0; m < MT; ++m) {
            // A-fragment: contiguous K pair -> single b64 LDS read
            v2f am = *(const v2f*)&s_mem[(m * 16 + laneM) * SM + k0];
            v2f aj; aj.x = am.x * mv[m]; aj.y = am.y * mv[m];   // masked copy for C-GEMM
            #pragma unroll
            for (int nt = 0; nt < 2; ++nt) {
                accA[m][nt] = __builtin_amdgcn_wmma_f32_16x16x4_f32(
                    false, am, false, bf1[nt], (short)0, accA[m][nt], false, false);
                accC[m][nt] = __builtin_amdgcn_wmma_f32_16x16x4_f32(
                    false, aj, false, bf2[nt], (short)0, accC[m][nt], false, false);
            }
        }
    }

    // D-matrix layout: lane holds (M = v + 8*laneH, N = laneM) -> scatter to LDS
    #pragma unroll
    for (int m = 0; m < MT; ++m)
        #pragma unroll
        for (int nt = 0; nt < 2; ++nt) {
            const int colD = colBase + nt * 16 + laneM;
            #pragma unroll
            for (int v = 0; v < 8; ++v) {
                const int rowD = m * 16 + 8 * laneH + v;
                s_a[rowD * SM + colD] = accA[m][nt][v];
                s_c[rowD * SM + colD] = accC[m][nt][v];
            }
        }
    __syncthreads();

    // -------- phase 2: gated off-diagonal reduction, 4 cols/thread (b128) --------
    for (int idx = tid; idx < ROWS * DDIM / 4; idx += NTHREADS) {   // 12 iters
        const int r    = idx >> 7;        // / 128
        const int col4 = (idx & 127) * 4;
        const int g    = r / NROW;
        const int i    = r - g * NROW;
        const int jb   = g * NROW;

        float4 av = *(const float4*)&s_a[r * SM + col4];
        float4 bv = *(const float4*)&s_bias[col4];
        float ab0 = av.x + bv.x, ab1 = av.y + bv.y, ab2 = av.z + bv.z, ab3 = av.w + bv.w;

        float a0 = 0.f, a1 = 0.f, a2 = 0.f, a3 = 0.f;
        #pragma unroll
        for (int j = 0; j < NROW; ++j) {
            if (j == i) continue;
            float4 cv = *(const float4*)&s_c[(jb + j) * SM + col4];
            float4 mv4 = *(const float4*)&s_mem[(jb + j) * SM + col4];
            const float mk = s_mask[jb + j];
            a0 += (mv4.x * mk) / (1.0f + __expf(-(ab0 + cv.x)));
            a1 += (mv4.y * mk) / (1.0f + __expf(-(ab1 + cv.y)));
            a2 += (mv4.z * mk) / (1.0f + __expf(-(ab2 + cv.z)));
            a3 += (mv4.w * mk) / (1.0f + __expf(-(ab3 + cv.w)));
        }
        float4 outv = make_float4(a0, a1, a2, a3);
        *(float4*)&out_g[(rowBase + r) * DDIM + col4] = outv;
    }
}

extern "C" void kernel_launch(void* const* d_in, const int* in_sizes, int n_in,
                              void* d_out, int out_size, void* d_ws, size_t ws_size,
                              hipStream_t stream)
{
    const float* memory = (const float*)d_in[0];   // (8192, 6, 512) f32
    const float* mask   = (const float*)d_in[1];   // (8192, 6, 1)  f32
    const float* W      = (const float*)d_in[2];   // (1024, 512)   f32
    const float* b      = (const float*)d_in[3];   // (512,)        f32
    float* out          = (float*)d_out;           // (8192, 6, 512) f32

    const int B = 8192;
    const int grid = B / GB;                       // 1024 workgroups
    context_fused_kernel<<<grid, NTHREADS, 0, stream>>>(memory, mask, W, b, out);
}